// SambaBlock_18202071400938
// MI455X (gfx1250) — compile-verified
//
#include <hip/hip_runtime.h>
#include <hip/hip_bf16.h>

// ---------------------------------------------------------------------------
// SambaBlock for MI455X (gfx1250).
// All matmuls run through a generic fp32 WMMA GEMM (V_WMMA_F32_16X16X4_F32).
// The op is HBM-bound on fp32 weights (~1.6GB/call -> ~70us @ 23.3TB/s), so:
//   - fp32 WMMA (no conversion traffic, full precision)
//   - grid mapped M-fastest so one weight tile is consumed by all M-blocks
//     back-to-back (weights stream from HBM once, activations stay in 192MB L2)
//   - 32-bit offsets from uniform bases -> global_load (saddr+voffset) only,
//     keeping the loop on LOADcnt (no flat/DScnt coupling)
//   - explicit global_prefetch_b8 scope:SCOPE_DEV pulls the weight stream
//     into L2 ahead of the WMMA consumer.
// ---------------------------------------------------------------------------

typedef float v2f __attribute__((ext_vector_type(2)));
typedef float v8f __attribute__((ext_vector_type(8)));

#define B_SZ 4
#define K_SZ 256
#define BK   (B_SZ * K_SZ)          // 1024 token rows
#define D_MODEL 256
#define D_STATE 16
#define D_INNER 512
#define DT_RANK 16
#define D_CONV  4
#define E_SZ   (D_INNER * D_STATE)  // 8192
#define FF_SZ  8192
#define H_SZ   8
#define DH     (E_SZ / H_SZ)        // 1024

// Prefetch one cacheline of the weight stream ~512 floats ahead into GL2.
#define GLOBAL_PF(p) \
    asm volatile("global_prefetch_b8 %0, off offset:2048 scope:SCOPE_DEV" :: "v"(p))

// ---------------------------------------------------------------------------
// Generic WMMA fp32 GEMM:  C[m,n] = alpha * sum_k A[m,k] * W[n*wsn + k*wsk]
//                                   (+ bias[n]) (+ R[m,n]) (relu?)
// Wave tile: 32(M) x 64(N). grid = (M/32, ceil(N/64), batch)  [M fastest!]
// CONTIG: W is contiguous in k (wsk==1). EDGE: N may not be a multiple of 64.
// ---------------------------------------------------------------------------
template <bool CONTIG, bool EDGE>
__global__ __launch_bounds__(32)
void gemm_f32_wmma(const float* __restrict__ A, const float* __restrict__ W,
                   const float* __restrict__ bias, const float* __restrict__ R,
                   float* __restrict__ C,
                   int N, int Kd, int lda, long long wsn, long long wsk, int ldc,
                   int zinner,
                   long long a_bo, long long a_bi,
                   long long w_bo, long long w_bi,
                   long long c_bo, long long c_bi,
                   float alpha, int relu)
{
    const int lane = threadIdx.x;          // blockDim.x == 32 (one wave)
    const int half = lane >> 4;            // 0: k{0,1}/rows 0..7, 1: k{2,3}/rows 8..15
    const int l16  = lane & 15;

    const int zo = (int)blockIdx.z / zinner;
    const int zi = (int)blockIdx.z % zinner;
    A += (long long)zo * a_bo + (long long)zi * a_bi;   // bases stay wave-uniform
    W += (long long)zo * w_bo + (long long)zi * w_bi;
    C += (long long)zo * c_bo + (long long)zi * c_bi;
    if (R) R += (long long)zo * c_bo + (long long)zi * c_bi;

    const int m0 = blockIdx.x * 32;        // M fastest for weight-tile L2 reuse
    const int n0 = blockIdx.y * 64;

    bool nv[4];
    int  ncol[4];
#pragma unroll
    for (int j = 0; j < 4; ++j) {
        ncol[j] = n0 + j * 16;
        nv[j]   = EDGE ? (ncol[j] < N) : true;
    }

    v8f acc[2][4];
#pragma unroll
    for (int mi = 0; mi < 2; ++mi)
#pragma unroll
        for (int j = 0; j < 4; ++j)
#pragma unroll
            for (int i = 0; i < 8; ++i)
                acc[mi][j][i] = 0.0f;

    const int ka = half * 2;                               // this lane's K sub-pair
    const unsigned wsku = (unsigned)wsk;

    // 32-bit element offsets from uniform bases -> global_load saddr+voffset form
    unsigned ao0 = (unsigned)(m0 + l16) * (unsigned)lda + (unsigned)ka;
    unsigned ao1 = ao0 + (unsigned)(16 * lda);
    unsigned wo[4];
#pragma unroll
    for (int j = 0; j < 4; ++j)
        wo[j] = (unsigned)((long long)(ncol[j] + l16) * wsn + (long long)ka * wsk);
    const unsigned wstep = 4u * wsku;

    // Dedicated prefetch pointers (kept off the load addressing path)
    const float* pf[4];
    if (CONTIG) {
#pragma unroll
        for (int j = 0; j < 4; ++j) pf[j] = W + wo[j];
    }

#pragma unroll 2
    for (int k = 0; k < Kd; k += 4) {
        v2f a0 = *(const v2f*)(A + ao0);
        v2f a1 = *(const v2f*)(A + ao1);
        v2f b[4];
#pragma unroll
        for (int j = 0; j < 4; ++j) {
            if (EDGE && !nv[j]) {
                b[j][0] = 0.0f; b[j][1] = 0.0f;
            } else if (CONTIG) {
                b[j] = *(const v2f*)(W + wo[j]);
                GLOBAL_PF(pf[j]);
                pf[j] += 4;
            } else {
                b[j][0] = W[wo[j]];
                b[j][1] = W[wo[j] + wsku];
            }
            wo[j] += wstep;
        }
        ao0 += 4;
        ao1 += 4;
#pragma unroll
        for (int mi = 0; mi < 2; ++mi) {
            v2f a = mi ? a1 : a0;
#pragma unroll
            for (int j = 0; j < 4; ++j) {
                acc[mi][j] = __builtin_amdgcn_wmma_f32_16x16x4_f32(
                    false, a, false, b[j], (short)0, acc[mi][j], false, false);
            }
        }
    }

    // Epilogue: D layout is VGPR r -> row (r + 8*half), col l16 within 16x16 tile.
#pragma unroll
    for (int mi = 0; mi < 2; ++mi) {
#pragma unroll
        for (int j = 0; j < 4; ++j) {
            if (EDGE && !nv[j]) continue;
            const int col = ncol[j] + l16;
            const float bv = bias ? bias[col] : 0.0f;
#pragma unroll
            for (int r = 0; r < 8; ++r) {
                const int row = m0 + mi * 16 + half * 8 + r;
                float v = acc[mi][j][r] * alpha + bv;
                if (R) v += R[(size_t)row * ldc + col];
                if (relu) v = fmaxf(v, 0.0f);
                C[(size_t)row * ldc + col] = v;
            }
        }
    }
}

// ---------------------------------------------------------------------------
// Elementwise / reduction kernels
// ---------------------------------------------------------------------------
__device__ __forceinline__ float silu_f(float x) { return x / (1.0f + expf(-x)); }

// conv history shift + depthwise conv + SiLU -> u ; writes new conv history to out
__global__ void conv_u_kernel(const float* __restrict__ xr,
                              const float* __restrict__ conv_in,
                              const float* __restrict__ conv_w,
                              const float* __restrict__ conv_b,
                              float* __restrict__ conv_out,
                              float* __restrict__ u)
{
    int idx = blockIdx.x * blockDim.x + threadIdx.x;     // bk*512 + d
    if (idx >= BK * D_INNER) return;
    int bk = idx >> 9, d = idx & 511;
    size_t cb = (size_t)bk * (D_CONV * D_INNER);
    float h0 = conv_in[cb + 1 * D_INNER + d];
    float h1 = conv_in[cb + 2 * D_INNER + d];
    float h2 = conv_in[cb + 3 * D_INNER + d];
    float h3 = xr[(size_t)bk * 1024 + d];                // x_in = first half of xr
    conv_out[cb + 0 * D_INNER + d] = h0;
    conv_out[cb + 1 * D_INNER + d] = h1;
    conv_out[cb + 2 * D_INNER + d] = h2;
    conv_out[cb + 3 * D_INNER + d] = h3;
    float acc = conv_b[d] + h0 * conv_w[d * 4 + 0] + h1 * conv_w[d * 4 + 1]
                          + h2 * conv_w[d * 4 + 2] + h3 * conv_w[d * 4 + 3];
    u[idx] = silu_f(acc);
}

// softplus(delta) + SSM state update -> x (= h flattened to (BK, 8192))
__global__ void ssm_kernel(const float* __restrict__ dtp,
                           const float* __restrict__ A_log,
                           const float* __restrict__ hidden,
                           const float* __restrict__ dbl,
                           const float* __restrict__ u,
                           float* __restrict__ x)
{
    int idx = blockIdx.x * blockDim.x + threadIdx.x;     // bk*8192 + d*16 + n
    if (idx >= BK * E_SZ) return;
    int bk = idx >> 13;
    int e  = idx & 8191;
    int d  = e >> 4;
    int n  = e & 15;
    float dt = dtp[(size_t)bk * D_INNER + d];
    float delta = (dt > 20.0f) ? dt : log1pf(expf(dt));
    float Aa = -expf(A_log[d * D_STATE + n]);
    float dA = expf(delta * Aa);
    float Bm = dbl[(size_t)bk * 48 + DT_RANK + n];
    x[idx] = dA * hidden[idx] + delta * Bm * u[(size_t)bk * D_INNER + d];
}

// softmax over last dim (256) of scores, in place; one 256-thread block per row
__global__ void softmax256_kernel(float* __restrict__ s)
{
    __shared__ float red[256];
    const size_t row = (size_t)blockIdx.x * 256;
    const int t = threadIdx.x;
    float v = s[row + t];
    red[t] = v; __syncthreads();
    for (int off = 128; off > 0; off >>= 1) {
        if (t < off) red[t] = fmaxf(red[t], red[t + off]);
        __syncthreads();
    }
    float mx = red[0]; __syncthreads();
    float e = expf(v - mx);
    red[t] = e; __syncthreads();
    for (int off = 128; off > 0; off >>= 1) {
        if (t < off) red[t] += red[t + off];
        __syncthreads();
    }
    s[row + t] = e / red[0];
}

// RMSNorm in place over rows of length n
__global__ void rmsnorm_kernel(float* __restrict__ x, const float* __restrict__ g, int n)
{
    __shared__ float red[256];
    const size_t base = (size_t)blockIdx.x * n;
    float ss = 0.0f;
    for (int i = threadIdx.x; i < n; i += 256) { float v = x[base + i]; ss += v * v; }
    red[threadIdx.x] = ss; __syncthreads();
    for (int off = 128; off > 0; off >>= 1) {
        if (threadIdx.x < off) red[threadIdx.x] += red[threadIdx.x + off];
        __syncthreads();
    }
    float scale = rsqrtf(red[0] / (float)n + 1e-6f);
    for (int i = threadIdx.x; i < n; i += 256) x[base + i] = x[base + i] * scale * g[i];
}

// y = h_new . Cm + u*Dp ; t = y * silu(res)
__global__ void yprep_kernel(const float* __restrict__ x3,
                             const float* __restrict__ dbl,
                             const float* __restrict__ u,
                             const float* __restrict__ Dp,
                             const float* __restrict__ xr,
                             float* __restrict__ t)
{
    int idx = blockIdx.x * blockDim.x + threadIdx.x;     // bk*512 + d
    if (idx >= BK * D_INNER) return;
    int bk = idx >> 9, d = idx & 511;
    const float* hb = x3 + (size_t)bk * E_SZ + (size_t)d * D_STATE;
    const float* Cm = dbl + (size_t)bk * 48 + DT_RANK + D_STATE;
    float acc = 0.0f;
#pragma unroll
    for (int n = 0; n < D_STATE; ++n) acc += hb[n] * Cm[n];
    float y = acc + u[idx] * Dp[d];
    float r = xr[(size_t)bk * 1024 + D_INNER + d];       // res = second half of xr
    t[idx] = y * silu_f(r);
}

// ---------------------------------------------------------------------------
// Host-side orchestration
// ---------------------------------------------------------------------------
static inline void launch_gemm(hipStream_t stream,
                               const float* A, const float* W, const float* bias,
                               const float* R, float* C,
                               int M, int N, int Kd, int lda, long long wsn, long long wsk,
                               int ldc, int batch, int zinner,
                               long long a_bo, long long a_bi,
                               long long w_bo, long long w_bi,
                               long long c_bo, long long c_bi,
                               float alpha, int relu)
{
    dim3 grid(M / 32, (N + 63) / 64, batch);   // M fastest -> weight L2 reuse
    const bool contig = (wsk == 1);
    const bool edge   = (N % 64) != 0;
    if (contig && !edge)
        gemm_f32_wmma<true, false><<<grid, 32, 0, stream>>>(A, W, bias, R, C, N, Kd, lda,
            wsn, wsk, ldc, zinner, a_bo, a_bi, w_bo, w_bi, c_bo, c_bi, alpha, relu);
    else if (contig && edge)
        gemm_f32_wmma<true, true><<<grid, 32, 0, stream>>>(A, W, bias, R, C, N, Kd, lda,
            wsn, wsk, ldc, zinner, a_bo, a_bi, w_bo, w_bi, c_bo, c_bi, alpha, relu);
    else if (!contig && !edge)
        gemm_f32_wmma<false, false><<<grid, 32, 0, stream>>>(A, W, bias, R, C, N, Kd, lda,
            wsn, wsk, ldc, zinner, a_bo, a_bi, w_bo, w_bi, c_bo, c_bi, alpha, relu);
    else
        gemm_f32_wmma<false, true><<<grid, 32, 0, stream>>>(A, W, bias, R, C, N, Kd, lda,
            wsn, wsk, ldc, zinner, a_bo, a_bi, w_bo, w_bi, c_bo, c_bi, alpha, relu);
}

extern "C" void kernel_launch(void* const* d_in, const int* in_sizes, int n_in,
                              void* d_out, int out_size, void* d_ws, size_t ws_size,
                              hipStream_t stream)
{
    const float* inputs     = (const float*)d_in[0];
    const float* hidden     = (const float*)d_in[1];
    const float* convh      = (const float*)d_in[2];
    const float* in_proj_w  = (const float*)d_in[3];
    const float* conv_w     = (const float*)d_in[4];
    const float* conv_b     = (const float*)d_in[5];
    const float* x_proj_w   = (const float*)d_in[6];
    const float* dt_proj_w  = (const float*)d_in[7];
    const float* dt_proj_b  = (const float*)d_in[8];
    const float* A_log      = (const float*)d_in[9];
    const float* Dp         = (const float*)d_in[10];
    const float* out_proj_w = (const float*)d_in[11];
    const float* qkv_w      = (const float*)d_in[12];
    const float* qkv_b      = (const float*)d_in[13];
    const float* attn_out_w = (const float*)d_in[14];
    const float* attn_out_b = (const float*)d_in[15];
    const float* norm1_g    = (const float*)d_in[16];
    const float* ffn_w1     = (const float*)d_in[17];
    const float* ffn_b1     = (const float*)d_in[18];
    const float* ffn_w2     = (const float*)d_in[19];
    const float* ffn_b2     = (const float*)d_in[20];
    const float* norm2_g    = (const float*)d_in[21];

    float* out       = (float*)d_out;                     // (BK,256)
    float* out_hnew  = out + (size_t)BK * D_MODEL;        // (BK,512,16)
    float* out_convh = out_hnew + (size_t)BK * E_SZ;      // (BK,4,512)

    float* ws = (float*)d_ws;
    float* xr     = ws;                                   // (BK,1024)
    float* u      = xr     + (size_t)BK * 1024;           // (BK,512)
    float* dbl    = u      + (size_t)BK * D_INNER;        // (BK,48)
    float* dtp    = dbl    + (size_t)BK * 48;             // (BK,512)
    float* x      = dtp    + (size_t)BK * D_INNER;        // (BK,8192)
    float* qkv    = x      + (size_t)BK * E_SZ;           // (BK,24576)
    float* scores = qkv    + (size_t)BK * 3 * E_SZ;       // (B,H,256,256)
    float* o      = scores + (size_t)B_SZ * H_SZ * K_SZ * K_SZ; // (BK,8192)
    float* x2     = o      + (size_t)BK * E_SZ;           // (BK,8192)
    float* x3     = x2     + (size_t)BK * E_SZ;           // (BK,8192)
    float* t      = x3     + (size_t)BK * E_SZ;           // (BK,512)
    float* f      = qkv;   // FFN hidden reuses qkv region (attention done by then)

    // 1) xr = inputs @ in_proj_w^T                        M=1024 N=1024 K=256
    launch_gemm(stream, inputs, in_proj_w, nullptr, nullptr, xr,
                BK, 1024, D_MODEL, D_MODEL, D_MODEL, 1, 1024,
                1, 1, 0, 0, 0, 0, 0, 0, 1.0f, 0);

    // 2) conv history shift + depthwise conv + SiLU -> u, new conv history to d_out
    conv_u_kernel<<<(BK * D_INNER + 255) / 256, 256, 0, stream>>>(
        xr, convh, conv_w, conv_b, out_convh, u);

    // 3) dbl = u @ x_proj_w^T                             M=1024 N=48 K=512
    launch_gemm(stream, u, x_proj_w, nullptr, nullptr, dbl,
                BK, 48, D_INNER, D_INNER, D_INNER, 1, 48,
                1, 1, 0, 0, 0, 0, 0, 0, 1.0f, 0);

    // 4) dtp = dt @ dt_proj_w^T + dt_proj_b               M=1024 N=512 K=16 (lda=48)
    launch_gemm(stream, dbl, dt_proj_w, dt_proj_b, nullptr, dtp,
                BK, D_INNER, DT_RANK, 48, DT_RANK, 1, D_INNER,
                1, 1, 0, 0, 0, 0, 0, 0, 1.0f, 0);

    // 5) SSM state update -> x = h.reshape(BK, 8192)
    ssm_kernel<<<(BK * E_SZ + 255) / 256, 256, 0, stream>>>(
        dtp, A_log, hidden, dbl, u, x);

    // 6) qkv = x @ qkv_w^T + qkv_b                        M=1024 N=24576 K=8192
    launch_gemm(stream, x, qkv_w, qkv_b, nullptr, qkv,
                BK, 3 * E_SZ, E_SZ, E_SZ, E_SZ, 1, 3 * E_SZ,
                1, 1, 0, 0, 0, 0, 0, 0, 1.0f, 0);

    // 7) scores[b,h] = (q @ k^T) / 32  (batched over 32 (b,h) pairs)
    launch_gemm(stream, qkv /*q*/, qkv + E_SZ /*k*/, nullptr, nullptr, scores,
                K_SZ, K_SZ, DH, 3 * E_SZ, 3 * E_SZ, 1, K_SZ,
                B_SZ * H_SZ, H_SZ,
                (long long)K_SZ * 3 * E_SZ, DH,
                (long long)K_SZ * 3 * E_SZ, DH,
                (long long)H_SZ * K_SZ * K_SZ, (long long)K_SZ * K_SZ,
                1.0f / 32.0f, 0);

    // 8) softmax over key dim
    softmax256_kernel<<<B_SZ * H_SZ * K_SZ, 256, 0, stream>>>(scores);

    // 9) o[b,h] = att @ v  (W strided: W[n,k] = v[k*24576 + n])
    launch_gemm(stream, scores, qkv + 2 * E_SZ /*v*/, nullptr, nullptr, o,
                K_SZ, DH, K_SZ, K_SZ, 1, 3 * E_SZ, E_SZ,
                B_SZ * H_SZ, H_SZ,
                (long long)H_SZ * K_SZ * K_SZ, (long long)K_SZ * K_SZ,
                (long long)K_SZ * 3 * E_SZ, DH,
                (long long)K_SZ * E_SZ, DH,
                1.0f, 0);

    // 10) x2 = x + o @ attn_out_w^T + attn_out_b          M=1024 N=8192 K=8192
    launch_gemm(stream, o, attn_out_w, attn_out_b, x, x2,
                BK, E_SZ, E_SZ, E_SZ, E_SZ, 1, E_SZ,
                1, 1, 0, 0, 0, 0, 0, 0, 1.0f, 0);

    // 11) rmsnorm(x2, norm1_g)
    rmsnorm_kernel<<<BK, 256, 0, stream>>>(x2, norm1_g, E_SZ);

    // 12) f = relu(x2 @ ffn_w1^T + ffn_b1)                M=1024 N=8192 K=8192
    launch_gemm(stream, x2, ffn_w1, ffn_b1, nullptr, f,
                BK, FF_SZ, E_SZ, E_SZ, E_SZ, 1, FF_SZ,
                1, 1, 0, 0, 0, 0, 0, 0, 1.0f, 1);

    // 13) x3 = x2 + f @ ffn_w2^T + ffn_b2                 M=1024 N=8192 K=8192
    launch_gemm(stream, f, ffn_w2, ffn_b2, x2, x3,
                BK, E_SZ, FF_SZ, FF_SZ, FF_SZ, 1, E_SZ,
                1, 1, 0, 0, 0, 0, 0, 0, 1.0f, 0);

    // 14) rmsnorm(x3, norm2_g)  -> x3 is h_new flattened
    rmsnorm_kernel<<<BK, 256, 0, stream>>>(x3, norm2_g, E_SZ);

    // 15) h_new output copy
    hipMemcpyAsync(out_hnew, x3, (size_t)BK * E_SZ * sizeof(float),
                   hipMemcpyDeviceToDevice, stream);

    // 16) t = (h_new . Cm + u*Dp) * silu(res)
    yprep_kernel<<<(BK * D_INNER + 255) / 256, 256, 0, stream>>>(
        x3, dbl, u, Dp, xr, t);

    // 17) out = t @ out_proj_w^T                          M=1024 N=256 K=512
    launch_gemm(stream, t, out_proj_w, nullptr, nullptr, out,
                BK, D_MODEL, D_INNER, D_INNER, D_INNER, 1, D_MODEL,
                1, 1, 0, 0, 0, 0, 0, 0, 1.0f, 0);
}